// AttentiveRNN_26792005992793
// MI455X (gfx1250) — compile-verified
//
#include <hip/hip_runtime.h>

typedef __attribute__((ext_vector_type(2))) float v2f;
typedef __attribute__((ext_vector_type(8))) float v8f;

#define TT   256
#define BB   512
#define DD   128
#define HH   50
#define KK   5
#define AA   4
#define HP   64      // H padded to 4 WMMA N-tiles
#define CH   104     // 2H=100 padded to multiple of 4
#define ROWS 16      // batch rows per workgroup (one WMMA M-tile)
#define NT   (TT + 1)
#define CHS  64      // ctx history stride (padded H)
#define KHS  8       // key history stride (padded K)

__launch_bounds__(256)
__global__ void arnn_kernel(const float* __restrict__ x,
                            const float* __restrict__ W_in,  const float* __restrict__ b_in,
                            const float* __restrict__ W_ctx, const float* __restrict__ b_ctx,
                            const float* __restrict__ W_key, const float* __restrict__ b_key,
                            const float* __restrict__ W_q,   const float* __restrict__ b_q,
                            const float* __restrict__ fc,
                            const float* __restrict__ W_act, const float* __restrict__ b_act,
                            float* __restrict__ out,
                            float* __restrict__ ctxh, float* __restrict__ keyh) {
  // ---- LDS state (~103 KB) ----
  __shared__ float sWinT[DD][HP];        // B-operand GEMM1: sWinT[k][n] = W_in[n][k]
  __shared__ float sWctxT[CH][HP];       // B-operand GEMM2: sWctxT[k][n] = W_ctx[n][k]
  __shared__ float sx[ROWS][DD];         // x tile for current step
  __shared__ float sa2[ROWS][CH];        // [prev_c | h | 0 pad]  (A-operand GEMM2)
  __shared__ float scv[ROWS][HP];        // c (current context), cols >= HH are 0
  __shared__ float ssc[ROWS][NT + 3];    // scores -> exp(p) per history step
  __shared__ float swb[ROWS][52];        // attention-weighted context w
  __shared__ float sq[ROWS][8];          // q
  __shared__ float sk[ROWS][8];          // key of current step
  __shared__ float sbin[HP], sbctx[HP];
  __shared__ float swkey[KK][52], swq[KK][52], swact[AA][52];
  __shared__ float sbkey[KK], sbq[KK], sbact[AA];
  __shared__ float srmax[ROWS], sden[ROWS];

  const int tid  = threadIdx.x;
  const int b0   = blockIdx.x * ROWS;
  const int wave = tid >> 5;
  const int lane = tid & 31;
  const int lm   = lane & 15;   // M (A/D) or N (B/D) index within tile
  const int lh   = lane >> 4;   // lane-half selector

  // ---- one-time staging: transpose + zero-pad weights into LDS ----
  for (int i = tid; i < DD * HP; i += 256) {
    int k = i >> 6, n = i & 63;
    sWinT[k][n] = (n < HH) ? W_in[n * DD + k] : 0.f;
  }
  for (int i = tid; i < CH * HP; i += 256) {
    int k = i >> 6, n = i & 63;
    sWctxT[k][n] = (n < HH && k < 2 * HH) ? W_ctx[n * (2 * HH) + k] : 0.f;
  }
  for (int i = tid; i < HP; i += 256) {
    sbin[i]  = (i < HH) ? b_in[i]  : 0.f;
    sbctx[i] = (i < HH) ? b_ctx[i] : 0.f;
  }
  for (int i = tid; i < KK * HH; i += 256) {
    int kk = i / HH, h = i % HH;
    swkey[kk][h] = W_key[i];
    swq[kk][h]   = W_q[i];
  }
  for (int i = tid; i < AA * HH; i += 256) {
    int a = i / HH, h = i % HH;
    swact[a][h] = W_act[i];
  }
  if (tid < KK) { sbkey[tid] = b_key[tid]; sbq[tid] = b_q[tid]; }
  if (tid < AA) { sbact[tid] = b_act[tid]; }
  // prev_c := first_context, concat pad cols [100..103] := 0 (stay 0 forever)
  for (int i = tid; i < ROWS * CH; i += 256) {
    int r = i / CH, j = i % CH;
    sa2[r][j] = (j < HH) ? fc[j] : 0.f;
  }
  __syncthreads();

  // ---- history init: ctx_hist[b][0][:] = fc ; key_hist[b][0][:] = fc@W_key^T + b_key ----
  for (int i = tid; i < ROWS * CHS; i += 256) {
    int r = i >> 6, h = i & 63;
    ctxh[((size_t)(b0 + r) * NT + 0) * CHS + h] = (h < HH) ? fc[h] : 0.f;
  }
  if (tid < ROWS * KK) {
    int r = tid / KK, kk = tid % KK;
    float acc = sbkey[kk];
    for (int h = 0; h < HH; ++h) acc += fc[h] * swkey[kk][h];
    keyh[((size_t)(b0 + r) * NT + 0) * KHS + kk] = acc;
  }
  __syncthreads();

  // ---- serial recurrence over T steps ----
  for (int t = 0; t < TT; ++t) {
    // phase 0: stage x_t tile
    for (int i = tid; i < ROWS * DD; i += 256) {
      int r = i >> 7, k = i & 127;
      sx[r][k] = x[((size_t)t * BB + (b0 + r)) * DD + k];
    }
    // prefetch next step's x tile into cache
    if (t + 1 < TT && tid < ROWS)
      __builtin_prefetch(&x[((size_t)(t + 1) * BB + (b0 + tid)) * DD], 0, 3);
    __syncthreads();

    // phase 1: GEMM1 via V_WMMA_F32_16X16X4_F32 : h = relu(x @ W_in^T + b_in)
    if (wave < 4) {
      v8f acc = {};
      const int n0 = wave * 16;
      for (int k0 = 0; k0 < DD; k0 += 4) {
        v2f a, b;
        a.x = sx[lm][k0 + 2 * lh];
        a.y = sx[lm][k0 + 2 * lh + 1];
        b.x = sWinT[k0 + 2 * lh][n0 + lm];
        b.y = sWinT[k0 + 2 * lh + 1][n0 + lm];
        acc = __builtin_amdgcn_wmma_f32_16x16x4_f32(false, a, false, b,
                                                    (short)0, acc, false, false);
      }
      const int n = n0 + lm;
      const float bv = sbin[n];
#pragma unroll
      for (int j = 0; j < 8; ++j) {
        int m = j + 8 * lh;
        float v = acc[j] + bv;
        v = v > 0.f ? v : 0.f;
        if (n < HH) sa2[m][HH + n] = v;   // h into concat buffer
      }
    }
    __syncthreads();

    // phase 2: GEMM2 via WMMA : c = relu([prev_c|h] @ W_ctx^T + b_ctx)
    if (wave < 4) {
      v8f acc = {};
      const int n0 = wave * 16;
      for (int k0 = 0; k0 < CH; k0 += 4) {
        v2f a, b;
        a.x = sa2[lm][k0 + 2 * lh];
        a.y = sa2[lm][k0 + 2 * lh + 1];
        b.x = sWctxT[k0 + 2 * lh][n0 + lm];
        b.y = sWctxT[k0 + 2 * lh + 1][n0 + lm];
        acc = __builtin_amdgcn_wmma_f32_16x16x4_f32(false, a, false, b,
                                                    (short)0, acc, false, false);
      }
      const int n = n0 + lm;
      const float bv = sbctx[n];
#pragma unroll
      for (int j = 0; j < 8; ++j) {
        int m = j + 8 * lh;
        float v = acc[j] + bv;
        v = v > 0.f ? v : 0.f;
        scv[m][n] = (n < HH) ? v : 0.f;
      }
    }
    __syncthreads();

    // phase 3: key/q projections (K=5), append history
    if (tid < ROWS * KK * 2) {
      const int which = (tid >= ROWS * KK);
      const int id = which ? tid - ROWS * KK : tid;
      const int r = id / KK, kk = id % KK;
      const float* wrow = which ? &swq[kk][0] : &swkey[kk][0];
      float acc = which ? sbq[kk] : sbkey[kk];
      for (int h = 0; h < HH; ++h) acc += scv[r][h] * wrow[h];
      if (which) {
        sq[r][kk] = acc;
      } else {
        sk[r][kk] = acc;
        keyh[((size_t)(b0 + r) * NT + (t + 1)) * KHS + kk] = acc;
      }
    }
    for (int i = tid; i < ROWS * CHS; i += 256) {
      int r = i >> 6, h = i & 63;
      ctxh[((size_t)(b0 + r) * NT + (t + 1)) * CHS + h] = scv[r][h];
    }
    if (tid < ROWS) sden[tid] = 0.f;
    __syncthreads();

    // phase 4: scores over full history (freshest entry from LDS)
    const int L = t + 2;
    for (int i = tid; i < ROWS * L; i += 256) {
      const int r = i & 15, tau = i >> 4;
      float s;
      if (tau == t + 1) {
        s = sk[r][0] * sq[r][0] + sk[r][1] * sq[r][1] + sk[r][2] * sq[r][2] +
            sk[r][3] * sq[r][3] + sk[r][4] * sq[r][4];
      } else {
        const float* kp = &keyh[((size_t)(b0 + r) * NT + tau) * KHS];
        s = kp[0] * sq[r][0] + kp[1] * sq[r][1] + kp[2] * sq[r][2] +
            kp[3] * sq[r][3] + kp[4] * sq[r][4];
      }
      ssc[r][tau] = s;
    }
    __syncthreads();

    // phase 5a: row max
    if (tid < ROWS) {
      float m = -1e30f;
      for (int tau = 0; tau < L; ++tau) m = fmaxf(m, ssc[tid][tau]);
      srmax[tid] = m;
    }
    __syncthreads();
    // phase 5b: exponentiate + denom (ds_add_f32 atomic)
    for (int i = tid; i < ROWS * L; i += 256) {
      const int r = i & 15, tau = i >> 4;
      const float p = __expf(ssc[r][tau] - srmax[r]);
      ssc[r][tau] = p;
      atomicAdd(&sden[r], p);
    }
    __syncthreads();

    // phase 6: w[r][h] = sum_tau p * ctx_hist (freshest term from LDS)
    for (int s = tid; s < ROWS * HH; s += 256) {
      const int r = s & 15, h = s >> 4;
      const float* cp = &ctxh[(size_t)(b0 + r) * NT * CHS + h];
      float acc = ssc[r][t + 1] * scv[r][h];
      for (int tau = 0; tau <= t; ++tau) acc += ssc[r][tau] * cp[(size_t)tau * CHS];
      swb[r][h] = acc / sden[r];
    }
    __syncthreads();

    // phase 7: action head (A=4)
    if (tid < ROWS * AA) {
      const int r = tid / AA, a = tid % AA;
      float acc = sbact[a];
      for (int h = 0; h < HH; ++h) acc += swb[r][h] * swact[a][h];
      out[((size_t)t * BB + (b0 + r)) * AA + a] = acc;
    }
    // phase 8: prev_c <- c
    for (int s = tid; s < ROWS * HH; s += 256) {
      const int r = s & 15, h = s >> 4;
      sa2[r][h] = scv[r][h];
    }
    __syncthreads();
  }
}

extern "C" void kernel_launch(void* const* d_in, const int* in_sizes, int n_in,
                              void* d_out, int out_size, void* d_ws, size_t ws_size,
                              hipStream_t stream) {
  const float* x     = (const float*)d_in[0];
  const float* W_in  = (const float*)d_in[1];
  const float* b_in  = (const float*)d_in[2];
  const float* W_ctx = (const float*)d_in[3];
  const float* b_ctx = (const float*)d_in[4];
  const float* W_key = (const float*)d_in[5];
  const float* b_key = (const float*)d_in[6];
  const float* W_q   = (const float*)d_in[7];
  const float* b_q   = (const float*)d_in[8];
  const float* fc    = (const float*)d_in[9];
  const float* W_act = (const float*)d_in[10];
  const float* b_act = (const float*)d_in[11];
  float* out = (float*)d_out;

  float* ctxh = (float*)d_ws;                       // [B][T+1][CHS]
  float* keyh = ctxh + (size_t)BB * NT * CHS;       // [B][T+1][KHS]

  dim3 grid(BB / ROWS), block(256);
  hipLaunchKernelGGL(arnn_kernel, grid, block, 0, stream,
                     x, W_in, b_in, W_ctx, b_ctx, W_key, b_key, W_q, b_q,
                     fc, W_act, b_act, out, ctxh, keyh);
}